// TruncatedConnection_21036749816197
// MI455X (gfx1250) — compile-verified
//
#include <hip/hip_runtime.h>
#include <stdint.h>

#define N_DATA  542080
#define N_TRUNC 40320
#define FEAT    64
#define TILE    256
#define WAVES   8

// ---------------------------------------------------------------------------
// Zero-fill (float4 grid-stride). d_out/d_ws are poisoned; atomics need zeros.
// ---------------------------------------------------------------------------
__global__ void zero_f4_kernel(float4* __restrict__ p, size_t n4) {
    size_t i      = (size_t)blockIdx.x * blockDim.x + threadIdx.x;
    size_t stride = (size_t)gridDim.x * blockDim.x;
    float4 z = make_float4(0.f, 0.f, 0.f, 0.f);
    for (; i < n4; i += stride) p[i] = z;
}

// ---------------------------------------------------------------------------
// Edge-weighted scatter: out[b, dst[e], :] += w[e] * h[b, src[e], :], b=0,1.
// Edge metadata staged into LDS via CDNA5 async global->LDS loads (ASYNCcnt).
// 16 lanes per edge: float4 (b128) covers 64 features; 2 edges per wave per
// iteration; both batches per edge reuse the staged metadata.
// ---------------------------------------------------------------------------
__global__ __launch_bounds__(256)
void edge_scatter_kernel(const float* __restrict__ h, size_t h_bstride,
                         const int*   __restrict__ src,
                         const int*   __restrict__ dst,
                         const float* __restrict__ w,
                         float* __restrict__ out, size_t out_bstride,
                         int n_edges) {
    __shared__ int   s_src[TILE];
    __shared__ int   s_dst[TILE];
    __shared__ float s_w[TILE];

    const int tid  = threadIdx.x;
    const int wave = tid >> 5;
    const int lane = tid & 31;
    const int half = lane >> 4;   // 0 or 1: which edge of the wave's pair
    const int l16  = lane & 15;   // 16 lanes x float4 = 64 features

    for (int base = (int)blockIdx.x * TILE; base < n_edges;
         base += (int)gridDim.x * TILE) {
        const int i = base + tid;

        __syncthreads();  // previous tile's LDS reads complete before overwrite

        if (i < n_edges) {
            // CDNA5 async path: global_load_async_to_lds_b32
            // vdst = per-lane LDS byte address, vaddr = 64-bit global address.
            uint32_t l0 = (uint32_t)(uintptr_t)&s_src[tid];
            uint32_t l1 = (uint32_t)(uintptr_t)&s_dst[tid];
            uint32_t l2 = (uint32_t)(uintptr_t)&s_w[tid];
            uint64_t g0 = (uint64_t)(uintptr_t)(src + i);
            uint64_t g1 = (uint64_t)(uintptr_t)(dst + i);
            uint64_t g2 = (uint64_t)(uintptr_t)(w + i);
            asm volatile("global_load_async_to_lds_b32 %0, %1, off"
                         :: "v"(l0), "v"(g0) : "memory");
            asm volatile("global_load_async_to_lds_b32 %0, %1, off"
                         :: "v"(l1), "v"(g1) : "memory");
            asm volatile("global_load_async_to_lds_b32 %0, %1, off"
                         :: "v"(l2), "v"(g2) : "memory");
        }
        asm volatile("s_wait_asynccnt 0" ::: "memory");
        __syncthreads();  // publish staged tile to all 8 waves

        #pragma unroll 4
        for (int it = 0; it < TILE / (WAVES * 2); ++it) {
            const int slot = it * (WAVES * 2) + wave * 2 + half;
            if (base + slot < n_edges) {
                const int   s  = s_src[slot];   // uniform per half-wave -> broadcast
                const int   d  = s_dst[slot];
                const float wt = s_w[slot];

                const float4* p0 =
                    (const float4*)(h + (size_t)s * FEAT) + l16;
                const float4* p1 =
                    (const float4*)(h + h_bstride + (size_t)s * FEAT) + l16;
                float4 v0 = *p0;   // coalesced global_load_b128, 256 B/half-wave
                float4 v1 = *p1;

                float* o0 = out + (size_t)d * FEAT + 4 * l16;
                float* o1 = o0 + out_bstride;
                unsafeAtomicAdd(o0 + 0, v0.x * wt);  // native global_atomic_add_f32
                unsafeAtomicAdd(o0 + 1, v0.y * wt);
                unsafeAtomicAdd(o0 + 2, v0.z * wt);
                unsafeAtomicAdd(o0 + 3, v0.w * wt);
                unsafeAtomicAdd(o1 + 0, v1.x * wt);
                unsafeAtomicAdd(o1 + 1, v1.y * wt);
                unsafeAtomicAdd(o1 + 2, v1.z * wt);
                unsafeAtomicAdd(o1 + 3, v1.w * wt);
            }
        }
    }
}

// ---------------------------------------------------------------------------
// Launcher: zero t and out, then down-projection, then up-projection.
// ---------------------------------------------------------------------------
extern "C" void kernel_launch(void* const* d_in, const int* in_sizes, int n_in,
                              void* d_out, int out_size, void* d_ws, size_t ws_size,
                              hipStream_t stream) {
    const float* x        = (const float*)d_in[0];
    const int*   down_src = (const int*)  d_in[1];
    const int*   down_dst = (const int*)  d_in[2];
    const float* down_w   = (const float*)d_in[3];
    const int*   up_src   = (const int*)  d_in[4];
    const int*   up_dst   = (const int*)  d_in[5];
    const float* up_w     = (const float*)d_in[6];

    float* out = (float*)d_out;
    float* t   = (float*)d_ws;  // intermediate: 2 * N_TRUNC * FEAT fp32 (~20.6 MB)

    const int e_down = in_sizes[1];
    const int e_up   = in_sizes[4];

    const size_t t_elems   = (size_t)2 * N_TRUNC * FEAT;
    const size_t out_elems = (size_t)2 * N_DATA  * FEAT;

    zero_f4_kernel<<<dim3(1024), dim3(256), 0, stream>>>((float4*)t,   t_elems / 4);
    zero_f4_kernel<<<dim3(4096), dim3(256), 0, stream>>>((float4*)out, out_elems / 4);

    // Stage 1: h = x[:, T-1, 0, :, :]  (batch b slice starts at (2b+1)*N*F)
    const float* h0        = x + (size_t)N_DATA * FEAT;       // b=0, T=1
    const size_t x_bstride = (size_t)2 * N_DATA * FEAT;       // to b=1, T=1

    int grid1 = (e_down + TILE - 1) / TILE;
    edge_scatter_kernel<<<dim3(grid1), dim3(256), 0, stream>>>(
        h0, x_bstride, down_src, down_dst, down_w,
        t, (size_t)N_TRUNC * FEAT, e_down);

    // Stage 2: t -> out
    int grid2 = (e_up + TILE - 1) / TILE;
    edge_scatter_kernel<<<dim3(grid2), dim3(256), 0, stream>>>(
        t, (size_t)N_TRUNC * FEAT, up_src, up_dst, up_w,
        out, (size_t)N_DATA * FEAT, e_up);
}